// SGC_41515153883312
// MI455X (gfx1250) — compile-verified
//
#include <hip/hip_runtime.h>
#include <hip/hip_bf16.h>

typedef __attribute__((ext_vector_type(2))) float v2f;
typedef __attribute__((ext_vector_type(8))) float v8f;

#define N_NODES   50000
#define N_EDGES   800000
#define N_FEAT    256
#define N_CLASSES 256
#define GROWS     80          // rows per GEMM block; 50000 = 625 * 80

// ------------------------------------------------------------- utilities ----
__global__ void k_zero(int* __restrict__ p, int n) {
    int i = blockIdx.x * 256 + threadIdx.x;
    if (i < n) p[i] = 0;
}

__global__ void k_cnt(const int* __restrict__ col, int* __restrict__ cnt) {
    int e = blockIdx.x * 256 + threadIdx.x;
    if (e < N_EDGES) atomicAdd(&cnt[col[e]], 1);
}

__global__ void k_dinv(const int* __restrict__ cnt, float* __restrict__ dinv) {
    int i = blockIdx.x * 256 + threadIdx.x;
    if (i < N_NODES) dinv[i] = rsqrtf((float)(cnt[i] + 1));   // +1 self-loop
}

// Single-block exclusive scan of cnt[0..N_NODES) -> rowPtr, rowPtr[N]=total.
__global__ void __launch_bounds__(1024) k_scan(const int* __restrict__ cnt,
                                               int* __restrict__ rowPtr) {
    __shared__ int part[1024];
    const int tid = threadIdx.x;
    const int CH  = 49;                       // 1024*49 = 50176 >= N_NODES
    const int base = tid * CH;
    int sum = 0;
    for (int j = 0; j < CH; ++j) {
        int idx = base + j;
        if (idx < N_NODES) sum += cnt[idx];
    }
    part[tid] = sum;
    __syncthreads();
    for (int off = 1; off < 1024; off <<= 1) {   // inclusive Hillis-Steele
        int v = (tid >= off) ? part[tid - off] : 0;
        __syncthreads();
        part[tid] += v;
        __syncthreads();
    }
    int run = (tid == 0) ? 0 : part[tid - 1];
    for (int j = 0; j < CH; ++j) {
        int idx = base + j;
        if (idx < N_NODES) { rowPtr[idx] = run; run += cnt[idx]; }
    }
    if (tid == 1023) rowPtr[N_NODES] = part[1023];
}

// Scatter edge sources into CSR-by-target slots.
__global__ void k_scatter(const int* __restrict__ row, const int* __restrict__ col,
                          const int* __restrict__ rowPtr, int* __restrict__ cur,
                          int* __restrict__ csrSrc) {
    int e = blockIdx.x * 256 + threadIdx.x;
    if (e >= N_EDGES) return;
    int c   = col[e];
    int pos = rowPtr[c] + atomicAdd(&cur[c], 1);
    csrSrc[pos] = row[e];
}

// --------------------------------------------- pull-style propagation hop ----
// hdst[v] = dinv[v]^2 * hsrc[v] + sum_{e: col=v} dinv[src]*dinv[v] * hsrc[src]
// One wave32 per target node: 8 feats/lane in registers, coalesced B128
// gathers from L2-resident hsrc, ONE plain coalesced store. No float atomics.
__global__ void k_hop_pull(const float* __restrict__ hsrc,
                           float*       __restrict__ hdst,
                           const int*   __restrict__ csrSrc,
                           const int*   __restrict__ rowPtr,
                           const float* __restrict__ dinv) {
    int node = blockIdx.x * 8 + (threadIdx.x >> 5);
    int lane = threadIdx.x & 31;
    if (node >= N_NODES) return;
    float dv = dinv[node];
    const float4* self = (const float4*)(hsrc + (size_t)node * N_FEAT);
    float4 a0 = self[lane], a1 = self[lane + 32];
    float s = dv * dv;
    a0.x *= s; a0.y *= s; a0.z *= s; a0.w *= s;
    a1.x *= s; a1.y *= s; a1.z *= s; a1.w *= s;
    int beg = rowPtr[node], end = rowPtr[node + 1];
    for (int i = beg; i < end; ++i) {
        int r   = csrSrc[i];
        float w = dinv[r] * dv;
        const float4* src = (const float4*)(hsrc + (size_t)r * N_FEAT);
        float4 v0 = src[lane], v1 = src[lane + 32];
        a0.x += v0.x * w; a0.y += v0.y * w; a0.z += v0.z * w; a0.w += v0.w * w;
        a1.x += v1.x * w; a1.y += v1.y * w; a1.z += v1.z * w; a1.w += v1.w * w;
    }
    float4* dst = (float4*)(hdst + (size_t)node * N_FEAT);
    dst[lane] = a0; dst[lane + 32] = a1;
}

// ------------------------------------------------------------ W transpose ---
__global__ void k_transpose(const float* __restrict__ W, float* __restrict__ Wt) {
    __shared__ float tile[32][33];
    int bx = blockIdx.x & 7, by = blockIdx.x >> 3;
    int tx = threadIdx.x & 31, ty = threadIdx.x >> 5;   // 32x8 threads
    #pragma unroll
    for (int i = 0; i < 32; i += 8)
        tile[ty + i][tx] = W[(by * 32 + ty + i) * N_FEAT + bx * 32 + tx];
    __syncthreads();
    #pragma unroll
    for (int i = 0; i < 32; i += 8)
        Wt[(bx * 32 + ty + i) * N_CLASSES + by * 32 + tx] = tile[tx][ty + i];
}

// ------------------------------------- fused WMMA GEMM + bias + logsoftmax --
// Block = 512 threads = 16 wave32s, one block per 80-row slab (625 blocks).
// Wave w owns columns [16w,16w+16) and 5 row-tiles: each B load pair is
// reused by 5 independent fp32 WMMAs (no D->A/B hazards). The 80x256 h tile
// in LDS is reused as the logits tile after a barrier.
__global__ void __launch_bounds__(512)
k_gemm_lsm(const float* __restrict__ h,     // aliases out; rows read first
           const float* __restrict__ Wt,    // [N_FEAT][N_CLASSES]
           const float* __restrict__ bias,  // [N_CLASSES]
           float*       __restrict__ out) { // [N_NODES][N_CLASSES]
    __shared__ float hTile[GROWS][260];
    const int tid     = threadIdx.x;
    const int rowBase = blockIdx.x * GROWS;

    // Stage 80x256 slab (5120 float4, 10 per thread), read BEFORE any write.
    {
        const float4* src = (const float4*)(h + (size_t)rowBase * N_FEAT);
        #pragma unroll
        for (int q = tid; q < GROWS * 64; q += 512) {
            float4 v = src[q];
            int r = q >> 6, c = (q & 63) << 2;
            hTile[r][c] = v.x; hTile[r][c + 1] = v.y;
            hTile[r][c + 2] = v.z; hTile[r][c + 3] = v.w;
        }
    }
    __syncthreads();

    const int wave = tid >> 5;
    const int lane = tid & 31;
    const int m    = lane & 15;
    const int kOff = (lane >> 4) << 1;      // 0 or 2
    const int nCol = wave * 16 + m;

    v8f acc[5] = {};
    const float* wtBase = Wt + nCol;
    for (int k0 = 0; k0 < N_FEAT; k0 += 4) {
        v2f b;
        b[0] = wtBase[(k0 + kOff) * N_CLASSES];
        b[1] = wtBase[(k0 + kOff + 1) * N_CLASSES];
        #pragma unroll
        for (int t = 0; t < 5; ++t) {
            v2f a;
            a[0] = hTile[t * 16 + m][k0 + kOff];
            a[1] = hTile[t * 16 + m][k0 + kOff + 1];
            acc[t] = __builtin_amdgcn_wmma_f32_16x16x4_f32(
                         false, a, false, b, (short)0, acc[t], false, false);
        }
    }
    __syncthreads();                         // all waves done reading hTile

    // Spill logits (+bias) into the reused LDS tile.
    float bv = bias[nCol];
    #pragma unroll
    for (int t = 0; t < 5; ++t)
        #pragma unroll
        for (int r = 0; r < 8; ++r)
            hTile[t * 16 + r + ((lane >> 4) << 3)][nCol] = acc[t][r] + bv;
    __syncthreads();

    // Row-wise log-softmax: wave w owns rows [5w, 5w+5), 8 elems/lane.
    for (int rr = 0; rr < 5; ++rr) {
        int row = wave * 5 + rr;
        const float* rowp = hTile[row];
        float v[8];
        float mx = -__builtin_inff();
        #pragma unroll
        for (int j = 0; j < 8; ++j) { v[j] = rowp[lane + 32 * j]; mx = fmaxf(mx, v[j]); }
        #pragma unroll
        for (int off = 16; off; off >>= 1) mx = fmaxf(mx, __shfl_xor(mx, off, 32));
        float s = 0.f;
        #pragma unroll
        for (int j = 0; j < 8; ++j) s += expf(v[j] - mx);
        #pragma unroll
        for (int off = 16; off; off >>= 1) s += __shfl_xor(s, off, 32);
        float lse = mx + logf(s);
        float* orow = out + (size_t)(rowBase + row) * N_CLASSES;
        #pragma unroll
        for (int j = 0; j < 8; ++j) orow[lane + 32 * j] = v[j] - lse;
    }
}

// ---------------------------------------------------------------- launch ----
extern "C" void kernel_launch(void* const* d_in, const int* in_sizes, int n_in,
                              void* d_out, int out_size, void* d_ws, size_t ws_size,
                              hipStream_t stream) {
    const float* x    = (const float*)d_in[0];                 // [50000,256]
    const int*   eidx = (const int*)d_in[1];                   // [2,800000]
    const float* W    = (const float*)d_in[2];                 // [256,256]
    const float* bias = (const float*)d_in[3];                 // [256]
    const int* row = eidx;                                     // sources
    const int* col = eidx + N_EDGES;                           // targets
    float* out = (float*)d_out;

    // Workspace layout (~55.9 MB total):
    char* ws = (char*)d_ws;
    int*   cnt    = (int*)  (ws + 0x000000);   // 200 KB
    int*   cur    = (int*)  (ws + 0x040000);   // 200 KB
    int*   rowPtr = (int*)  (ws + 0x080000);   // 200 KB (+1)
    float* dinv   = (float*)(ws + 0x0C0000);   // 200 KB
    float* Wt     = (float*)(ws + 0x100000);   // 256 KB
    int*   csrSrc = (int*)  (ws + 0x140000);   // 3.2 MB
    float* h1     = (float*)(ws + 0x480000);   // 51.2 MB

    const int NB_N = (N_NODES + 255) / 256;
    const int NB_E = N_EDGES / 256;            // exact

    // CSR-by-target construction + normalization.
    k_zero   <<<NB_N, 256, 0, stream>>>(cnt, N_NODES);
    k_cnt    <<<NB_E, 256, 0, stream>>>(col, cnt);
    k_dinv   <<<NB_N, 256, 0, stream>>>(cnt, dinv);
    k_scan   <<<1, 1024, 0, stream>>>(cnt, rowPtr);
    k_zero   <<<NB_N, 256, 0, stream>>>(cur, N_NODES);
    k_scatter<<<NB_E, 256, 0, stream>>>(row, col, rowPtr, cur, csrSrc);
    k_transpose<<<64, 256, 0, stream>>>(W, Wt);

    // Two pull-style hops (atomic-free; h buffers are L2-resident).
    k_hop_pull<<<N_NODES / 8, 256, 0, stream>>>(x,  h1,  csrSrc, rowPtr, dinv);
    k_hop_pull<<<N_NODES / 8, 256, 0, stream>>>(h1, out, csrSrc, rowPtr, dinv);

    // Fused fp32-WMMA linear + log_softmax, in-place on out.
    k_gemm_lsm<<<N_NODES / GROWS, 512, 0, stream>>>(out, Wt, bias, out);
}